// MixHop_35588099015579
// MI455X (gfx1250) — compile-verified
//
#include <hip/hip_runtime.h>
#include <math.h>

typedef __attribute__((ext_vector_type(2))) float v2f;
typedef __attribute__((ext_vector_type(8))) float v8f;

#define F_IN 128
#define HID  64
#define PPOW 3
#define BN_EPS 1e-5f

// ----------------------------------------------------------------------------
// Degree / symmetric normalization
// ----------------------------------------------------------------------------
__global__ void k_init_deg(float* __restrict__ deg, int n) {
  int i = blockIdx.x * blockDim.x + threadIdx.x;
  if (i < n) deg[i] = 1.0f;  // self-loop contributes 1
}

__global__ void k_deg(const long long* __restrict__ dst, float* __restrict__ deg, int e) {
  int i = blockIdx.x * blockDim.x + threadIdx.x;
  if (i < e) atomicAdd(&deg[dst[i]], 1.0f);
}

__global__ void k_dinv(const float* __restrict__ deg, float* __restrict__ dinv, int n) {
  int i = blockIdx.x * blockDim.x + threadIdx.x;
  if (i < n) {
    float d = deg[i];
    dinv[i] = (d > 0.0f) ? rsqrtf(d) : 0.0f;
  }
}

// ----------------------------------------------------------------------------
// SpMM: out = Ahat * in   (Ahat = D^-1/2 (A+I) D^-1/2), F multiple of 4
// Self-loop term initializes out (plain store), then edge scatter-adds.
// ----------------------------------------------------------------------------
__global__ void k_selfloop(const float* __restrict__ in, const float* __restrict__ dinv,
                           float* __restrict__ out, int n, int F) {
  int chunks = F >> 2;
  long long t = (long long)blockIdx.x * blockDim.x + threadIdx.x;
  if (t >= (long long)n * chunks) return;
  int row = (int)(t / chunks);
  int c   = (int)(t % chunks) << 2;
  float w = dinv[row] * dinv[row];
  float4 v = *(const float4*)(in + (long long)row * F + c);
  float4 o;
  o.x = v.x * w; o.y = v.y * w; o.z = v.z * w; o.w = v.w * w;
  *(float4*)(out + (long long)row * F + c) = o;
}

__global__ void k_spmm_edges(const long long* __restrict__ src, const long long* __restrict__ dst,
                             const float* __restrict__ dinv, const float* __restrict__ in,
                             float* __restrict__ out, int e, int F) {
  int chunks = F >> 2;  // 32 for F=128 (one wave/edge), 48 for F=192
  long long t = (long long)blockIdx.x * blockDim.x + threadIdx.x;
  if (t >= (long long)e * chunks) return;
  int ed = (int)(t / chunks);
  int c  = (int)(t % chunks) << 2;
  long long s = src[ed];
  long long d = dst[ed];
  float w = dinv[s] * dinv[d];
  float4 v = *(const float4*)(in + s * F + c);  // coalesced 512B gather per edge (L2-resident)
  float* o = out + d * F + c;
  atomicAdd(o + 0, v.x * w);
  atomicAdd(o + 1, v.y * w);
  atomicAdd(o + 2, v.z * w);
  atomicAdd(o + 3, v.w * w);
}

// ----------------------------------------------------------------------------
// GEMM via V_WMMA_F32_16X16X4_F32:  out[n x 64] = A[n x F] @ W[F x 64] + bias
// Block = 128 threads = 4 wave32; wave w -> 16x16 tile at cols [16w, 16w+16).
// A tile (16 x F) staged in LDS with +4 padding to avoid bank conflicts.
// Staging uses CDNA5 async memory->LDS DMA (GLOBAL_LOAD_ASYNC_TO_LDS_B128,
// tracked with ASYNCcnt) -- no VGPR round-trip for the tile fill. Dynamic
// shared memory is the only LDS allocation, so its byte offsets are valid
// LDS addresses for the async unit.
// out is written with stride `ostride` (pre-offset for concat column block).
// ----------------------------------------------------------------------------
__global__ void k_gemm_wmma(const float* __restrict__ A, const float* __restrict__ W,
                            const float* __restrict__ bias, float* __restrict__ out,
                            int n, int F, int ostride) {
  extern __shared__ float sA[];       // 16 * (F + 4)
  const int ldS  = F + 4;
  const int row0 = blockIdx.x * 16;
  const int tid  = threadIdx.x;

  // async stage of the 16 x F A-tile: one 16B chunk per lane per iteration
  const int chunksA = F >> 2;
  for (int idx = tid; idx < 16 * chunksA; idx += blockDim.x) {
    int r = idx / chunksA;
    int c = (idx - r * chunksA) << 2;
    int gr = row0 + r;
    unsigned lds_off = (unsigned)((r * ldS + c) * sizeof(float));
    if (gr < n) {
      const float* gp = A + (long long)gr * F + c;
      asm volatile("global_load_async_to_lds_b128 %0, %1, off"
                   :: "v"(lds_off), "v"(gp)
                   : "memory");
    } else {
      *(float4*)(&sA[r * ldS + c]) = make_float4(0.f, 0.f, 0.f, 0.f);
    }
  }

  // speculative prefetch of the next block's A tile into L2 (linear walk)
  if (row0 + 16 < n) {
    __builtin_prefetch(A + (long long)(row0 + 16) * F + tid * 4, 0, 1);
  }

  asm volatile("s_wait_asynccnt 0x0" ::: "memory");
  __syncthreads();

  const int wave  = tid >> 5;
  const int lane  = tid & 31;
  const int rsel  = lane & 15;   // M for A-frag, N for B/C/D-frag
  const int khalf = lane >> 4;   // 0: K 0..1, 1: K 2..3 within the 4-step
  const int gcol  = wave * 16 + rsel;

  v8f acc = {};
  for (int k0 = 0; k0 < F; k0 += 4) {
    int ka = k0 + 2 * khalf;
    v2f a, b;
    a.x = sA[rsel * ldS + ka];
    a.y = sA[rsel * ldS + ka + 1];
    b.x = W[(long long)ka * HID + gcol];
    b.y = W[(long long)(ka + 1) * HID + gcol];
    acc = __builtin_amdgcn_wmma_f32_16x16x4_f32(
        /*neg_a=*/false, a, /*neg_b=*/false, b,
        /*c_mod=*/(short)0, acc, /*reuse_a=*/false, /*reuse_b=*/false);
  }

  const float bv = bias[gcol];
  const int rbase = row0 + (khalf ? 8 : 0);   // C/D layout: lanes 16-31 hold M=8..15
#pragma unroll
  for (int r = 0; r < 8; ++r) {
    int grow = rbase + r;
    if (grow < n) out[(long long)grow * ostride + gcol] = acc[r] + bv;
  }
}

// ----------------------------------------------------------------------------
// BatchNorm (training mode, biased variance) + ReLU
// ----------------------------------------------------------------------------
__global__ void k_zero(float* __restrict__ p, int n) {
  int i = blockIdx.x * blockDim.x + threadIdx.x;
  if (i < n) p[i] = 0.0f;
}

// blockDim.x == C (192 = 6 waves); each block reduces a row-chunk, then column atomics
__global__ void k_bn_stats(const float* __restrict__ h, float* __restrict__ sum,
                           float* __restrict__ sumsq, int n, int C, int rows_per_block) {
  int col = threadIdx.x;
  int r0 = blockIdx.x * rows_per_block;
  int r1 = r0 + rows_per_block;
  if (r1 > n) r1 = n;
  float s = 0.0f, q = 0.0f;
  for (int r = r0; r < r1; ++r) {
    float v = h[(long long)r * C + col];
    s += v;
    q += v * v;
  }
  atomicAdd(&sum[col], s);
  atomicAdd(&sumsq[col], q);
}

__global__ void k_bn_final(const float* __restrict__ sum, const float* __restrict__ sumsq,
                           const float* __restrict__ g, const float* __restrict__ be,
                           float* __restrict__ scale, float* __restrict__ shift, int n, int C) {
  int c = blockIdx.x * blockDim.x + threadIdx.x;
  if (c >= C) return;
  float inv_n = 1.0f / (float)n;
  float mu  = sum[c] * inv_n;
  float var = sumsq[c] * inv_n - mu * mu;
  float a = g[c] * rsqrtf(var + BN_EPS);
  scale[c] = a;
  shift[c] = be[c] - mu * a;
}

__global__ void k_bn_relu(float* __restrict__ h, const float* __restrict__ scale,
                          const float* __restrict__ shift, long long total, int C) {
  long long i = (long long)blockIdx.x * blockDim.x + threadIdx.x;
  if (i >= total) return;
  int c = (int)(i % C);
  float v = h[i] * scale[c] + shift[c];
  h[i] = (v > 0.0f) ? v : 0.0f;
}

// ----------------------------------------------------------------------------
// Launcher
// ----------------------------------------------------------------------------
static inline unsigned grid1(long long total, int block) {
  return (unsigned)((total + block - 1) / block);
}

extern "C" void kernel_launch(void* const* d_in, const int* in_sizes, int n_in,
                              void* d_out, int out_size, void* d_ws, size_t ws_size,
                              hipStream_t stream) {
  const float*      x   = (const float*)d_in[0];
  const long long*  ei  = (const long long*)d_in[1];   // (2, E) int64
  const float*      W1  = (const float*)d_in[2];       // (3, 128, 64)
  const float*      B1  = (const float*)d_in[3];       // (3, 64)
  const float*      g1  = (const float*)d_in[4];       // (192,)
  const float*      be1 = (const float*)d_in[5];
  const float*      W2  = (const float*)d_in[6];       // (3, 192, 64)
  const float*      B2  = (const float*)d_in[7];       // (3, 64)
  const float*      g2  = (const float*)d_in[8];
  const float*      be2 = (const float*)d_in[9];
  const float*      Wfc = (const float*)d_in[10];      // (192, 64)
  const float*      bfc = (const float*)d_in[11];      // (64,)
  float*            out = (float*)d_out;

  const int N = in_sizes[0] / F_IN;
  const int E = in_sizes[1] / 2;
  const int C = PPOW * HID;                            // 192
  const long long NC = (long long)N * C;

  const long long* src = ei;
  const long long* dst = ei + E;

  // workspace carve-up (fp32)
  float* ws    = (float*)d_ws;
  float* deg   = ws;                 ws += N;
  float* dinv  = ws;                 ws += N;
  float* x1    = ws;                 ws += (long long)N * F_IN;
  float* x2    = ws;                 ws += (long long)N * F_IN;
  float* h1    = ws;                 ws += NC;
  float* y1    = ws;                 ws += NC;
  float* y2    = ws;                 ws += NC;
  float* h2    = ws;                 ws += NC;
  float* ssum  = ws;                 ws += C;
  float* ssq   = ws;                 ws += C;
  float* scl   = ws;                 ws += C;
  float* shf   = ws;                 ws += C;

  const int blk = 256;
  const unsigned gemmGrid = (unsigned)((N + 15) / 16);
  const size_t shm128 = (size_t)16 * (F_IN + 4) * sizeof(float);
  const size_t shm192 = (size_t)16 * (C + 4) * sizeof(float);
  const int rows_pb = 256;

  // ---- normalization coefficients ----
  k_init_deg<<<grid1(N, blk), blk, 0, stream>>>(deg, N);
  k_deg<<<grid1(E, blk), blk, 0, stream>>>(dst, deg, E);
  k_dinv<<<grid1(N, blk), blk, 0, stream>>>(deg, dinv, N);

  // ---- layer 1 (F = 128) ----
  k_gemm_wmma<<<gemmGrid, 128, shm128, stream>>>(x, W1 + 0 * F_IN * HID, B1 + 0 * HID,
                                                 h1 + 0 * HID, N, F_IN, C);
  k_selfloop<<<grid1((long long)N * (F_IN / 4), blk), blk, 0, stream>>>(x, dinv, x1, N, F_IN);
  k_spmm_edges<<<grid1((long long)E * (F_IN / 4), blk), blk, 0, stream>>>(src, dst, dinv, x, x1, E, F_IN);
  k_gemm_wmma<<<gemmGrid, 128, shm128, stream>>>(x1, W1 + 1 * F_IN * HID, B1 + 1 * HID,
                                                 h1 + 1 * HID, N, F_IN, C);
  k_selfloop<<<grid1((long long)N * (F_IN / 4), blk), blk, 0, stream>>>(x1, dinv, x2, N, F_IN);
  k_spmm_edges<<<grid1((long long)E * (F_IN / 4), blk), blk, 0, stream>>>(src, dst, dinv, x1, x2, E, F_IN);
  k_gemm_wmma<<<gemmGrid, 128, shm128, stream>>>(x2, W1 + 2 * F_IN * HID, B1 + 2 * HID,
                                                 h1 + 2 * HID, N, F_IN, C);

  // ---- BN1 + ReLU ----
  k_zero<<<grid1(2 * C, blk), blk, 0, stream>>>(ssum, 2 * C);   // ssum & ssq contiguous
  k_bn_stats<<<grid1(N, rows_pb), C, 0, stream>>>(h1, ssum, ssq, N, C, rows_pb);
  k_bn_final<<<1, C, 0, stream>>>(ssum, ssq, g1, be1, scl, shf, N, C);
  k_bn_relu<<<grid1(NC, blk), blk, 0, stream>>>(h1, scl, shf, NC, C);

  // ---- layer 2 (F = 192) ----
  k_gemm_wmma<<<gemmGrid, 128, shm192, stream>>>(h1, W2 + 0 * C * HID, B2 + 0 * HID,
                                                 h2 + 0 * HID, N, C, C);
  k_selfloop<<<grid1((long long)N * (C / 4), blk), blk, 0, stream>>>(h1, dinv, y1, N, C);
  k_spmm_edges<<<grid1((long long)E * (C / 4), blk), blk, 0, stream>>>(src, dst, dinv, h1, y1, E, C);
  k_gemm_wmma<<<gemmGrid, 128, shm192, stream>>>(y1, W2 + 1 * C * HID, B2 + 1 * HID,
                                                 h2 + 1 * HID, N, C, C);
  k_selfloop<<<grid1((long long)N * (C / 4), blk), blk, 0, stream>>>(y1, dinv, y2, N, C);
  k_spmm_edges<<<grid1((long long)E * (C / 4), blk), blk, 0, stream>>>(src, dst, dinv, y1, y2, E, C);
  k_gemm_wmma<<<gemmGrid, 128, shm192, stream>>>(y2, W2 + 2 * C * HID, B2 + 2 * HID,
                                                 h2 + 2 * HID, N, C, C);

  // ---- BN2 + ReLU ----
  k_zero<<<grid1(2 * C, blk), blk, 0, stream>>>(ssum, 2 * C);
  k_bn_stats<<<grid1(N, rows_pb), C, 0, stream>>>(h2, ssum, ssq, N, C, rows_pb);
  k_bn_final<<<1, C, 0, stream>>>(ssum, ssq, g2, be2, scl, shf, N, C);
  k_bn_relu<<<grid1(NC, blk), blk, 0, stream>>>(h2, scl, shf, NC, C);

  // ---- final FC: out = h2 @ Wfc + bfc ----
  k_gemm_wmma<<<gemmGrid, 128, shm192, stream>>>(h2, Wfc, bfc, out, N, C, HID);
}